// FakeFlexOlmoExperts_11793980194916
// MI455X (gfx1250) — compile-verified
//
#include <hip/hip_runtime.h>
#include <hip/hip_bf16.h>

// ---------------------------------------------------------------------------
// FakeFlexOlmo MoE experts, MI455X (gfx1250):
//   routed (4x fewer FLOPs than dense reference), fp32 WMMA matrix pipe,
//   fused 2-layer FFN, 64-token tile with X/h1 aliased in the 320KB WGP LDS.
//   Per k-step per wave: 4 ds_load_b64 + 8 global_load_b32 + 16 v_wmma.
// ---------------------------------------------------------------------------

#define B_ 4
#define S_ 2048
#define H_ 1024
#define E_ 8
#define K_ 2
#define N_ (B_ * S_)          // 8192 tokens
#define CAP_ (N_ * K_)        // 16384 max entries per expert (provable bound)
#define TM_ 64                // tokens per tile
#define MT_ 4                 // 16-row M tiles per wave (TM_/16)
#define NT_ 4                 // 16-col N tiles per wave
#define WAVES_ 16
#define BLOCK_ (WAVES_ * 32)  // 512 threads
#define LDK_ 1028             // padded LDS row stride (conflict-free ds_load_b64)
#define TILES_ (CAP_ / TM_)   // 256

typedef float v2f __attribute__((ext_vector_type(2)));
typedef float v8f __attribute__((ext_vector_type(8)));

__device__ __forceinline__ v8f wmma_f32(v2f a, v2f b, v8f c) {
  // D = A(16x4 f32) x B(4x16 f32) + C(16x16 f32)
  return __builtin_amdgcn_wmma_f32_16x16x4_f32(
      /*neg_a=*/false, a, /*neg_b=*/false, b,
      /*c_mod=*/(short)0, c, /*reuse_a=*/false, /*reuse_b=*/false);
}

// acc[mt][nt] = As(64 x 1024, LDS) @ Bg(1024 x 1024, global) + bias
// Wave owns 64 output columns starting at n0; lhalf/l16 per ISA fragment layout.
__device__ __forceinline__ void gemm_phase(const float* __restrict__ As,
                                           const float* __restrict__ Bg,
                                           const float* __restrict__ bias,
                                           int n0, int lhalf, int l16,
                                           v8f acc[MT_][NT_]) {
  // Init accumulators with bias: every C VGPR of a lane holds the same N column.
#pragma unroll
  for (int nt = 0; nt < NT_; ++nt) {
    float bv = bias[n0 + nt * 16 + l16];
    v8f c;
#pragma unroll
    for (int v = 0; v < 8; ++v) c[v] = bv;
#pragma unroll
    for (int mt = 0; mt < MT_; ++mt) acc[mt][nt] = c;
  }
  for (int kk = 0; kk < H_; kk += 4) {
    const int acol = kk + 2 * lhalf;  // K pair owned by this lane-half
    v2f a[MT_];
#pragma unroll
    for (int mt = 0; mt < MT_; ++mt)  // conflict-free ds_load_b64 (row pad +4)
      a[mt] = *(const v2f*)&As[(mt * 16 + l16) * LDK_ + acol];
#pragma unroll
    for (int nt = 0; nt < NT_; ++nt) {
      const float* wp = Bg + (size_t)acol * H_ + (n0 + nt * 16 + l16);
      v2f bf;
      bf[0] = wp[0];    // B[acol    ][col]
      bf[1] = wp[H_];   // B[acol + 1][col]
#pragma unroll
      for (int mt = 0; mt < MT_; ++mt)  // one B fragment feeds 4 WMMAs
        acc[mt][nt] = wmma_f32(a[mt], bf, acc[mt][nt]);
    }
  }
}

// ---------------------------------------------------------------------------
__global__ void zero_kernel(float* __restrict__ out, int* __restrict__ counts,
                            size_t n) {
  size_t i = (size_t)blockIdx.x * blockDim.x + threadIdx.x;
  size_t stride = (size_t)gridDim.x * blockDim.x;
  for (; i < n; i += stride) out[i] = 0.0f;
  if (blockIdx.x == 0 && threadIdx.x < E_) counts[threadIdx.x] = 0;
}

__global__ void route_kernel(const long long* __restrict__ tki,
                             const float* __restrict__ tkw,
                             int* __restrict__ counts,
                             int* __restrict__ tok,
                             float* __restrict__ twg) {
  int t = blockIdx.x * blockDim.x + threadIdx.x;
  if (t >= N_) return;
#pragma unroll
  for (int k = 0; k < K_; ++k) {
    int e = (int)tki[(size_t)t * K_ + k];
    float w = tkw[(size_t)t * K_ + k];
    int pos = atomicAdd(&counts[e], 1);
    tok[(size_t)e * CAP_ + pos] = t;
    twg[(size_t)e * CAP_ + pos] = w;
  }
}

// Fused 2-layer expert FFN over a 64-token gathered tile.
// 16 waves; wave w owns output columns [64w, 64w+64).
__global__ __launch_bounds__(BLOCK_, 1) void moe_expert_gemm(
    const float* __restrict__ X,    // [N,H]
    const float* __restrict__ w1,   // [E,H,H]  (k-major: w1[e][k][n])
    const float* __restrict__ b1,   // [E,H]
    const float* __restrict__ w2,   // [E,H,H]
    const float* __restrict__ b2,   // [E,H]
    const int* __restrict__ counts,
    const int* __restrict__ tok,
    const float* __restrict__ twg,
    float* __restrict__ out) {      // [N,H]
  const int e = blockIdx.y;
  const int cnt = min(counts[e], CAP_);
  const int row0 = blockIdx.x * TM_;
  if (row0 >= cnt) return;  // uniform per block: EXEC stays all-ones for WMMA
  const int nrows = min(TM_, cnt - row0);

  extern __shared__ char smem_raw[];
  float* Ts = (float*)smem_raw;        // [TM_][LDK_]  X tile, then relu(h1) tile
  int* tids = (int*)(Ts + TM_ * LDK_); // [TM_]
  float* tws = (float*)(tids + TM_);   // [TM_]

  const int tid = threadIdx.x;
  if (tid < TM_) {
    if (tid < nrows) {
      tids[tid] = tok[(size_t)e * CAP_ + row0 + tid];
      tws[tid] = twg[(size_t)e * CAP_ + row0 + tid];
    } else {
      tids[tid] = -1;
      tws[tid] = 0.0f;
    }
  }
  __syncthreads();

  // Gather X tile into LDS (float4, coalesced; padded rows stay 16B-aligned).
  for (int i = tid; i < TM_ * (H_ / 4); i += BLOCK_) {
    int r = i >> 8;        // H_/4 == 256 float4 per row
    int c4 = i & 255;
    float4 v = make_float4(0.f, 0.f, 0.f, 0.f);
    if (r < nrows) v = ((const float4*)(X + (size_t)tids[r] * H_))[c4];
    *(float4*)&Ts[r * LDK_ + c4 * 4] = v;
  }
  __syncthreads();

  const int lane = tid & 31;
  const int wave = tid >> 5;
  const int lhalf = lane >> 4;  // which K-pair / M+8 half this lane holds
  const int l16 = lane & 15;    // N column within 16-wide tile
  const int n0 = wave * (NT_ * 16);

  const float* w1e = w1 + (size_t)e * H_ * H_;
  const float* w2e = w2 + (size_t)e * H_ * H_;

  v8f acc[MT_][NT_];

  // ---- layer 1: h1 = relu(X @ w1[e] + b1[e]) ----
  gemm_phase(Ts, w1e, b1 + (size_t)e * H_, n0, lhalf, l16, acc);
  __syncthreads();  // all waves done reading X before h1 overwrites the buffer
#pragma unroll
  for (int nt = 0; nt < NT_; ++nt) {
    int col = n0 + nt * 16 + l16;
#pragma unroll
    for (int mt = 0; mt < MT_; ++mt) {
#pragma unroll
      for (int v = 0; v < 8; ++v) {
        int m = mt * 16 + v + 8 * lhalf;  // C layout: VGPR v -> rows v / v+8
        float x = acc[mt][nt][v];
        Ts[m * LDK_ + col] = x > 0.0f ? x : 0.0f;
      }
    }
  }
  __syncthreads();

  // ---- layer 2: h2 = h1 @ w2[e] + b2[e]; out[t] += tkw * h2 ----
  gemm_phase(Ts, w2e, b2 + (size_t)e * H_, n0, lhalf, l16, acc);
#pragma unroll
  for (int mt = 0; mt < MT_; ++mt) {
#pragma unroll
    for (int v = 0; v < 8; ++v) {
      int m = mt * 16 + v + 8 * lhalf;
      if (m < nrows) {
        int t = tids[m];
        float wv = tws[m];
        float* op = out + (size_t)t * H_ + n0 + l16;
#pragma unroll
        for (int nt = 0; nt < NT_; ++nt)
          unsafeAtomicAdd(op + nt * 16, wv * acc[mt][nt][v]);  // global_atomic_add_f32
      }
    }
  }
}

// ---------------------------------------------------------------------------
extern "C" void kernel_launch(void* const* d_in, const int* in_sizes, int n_in,
                              void* d_out, int out_size, void* d_ws,
                              size_t ws_size, hipStream_t stream) {
  const float* X = (const float*)d_in[0];
  const long long* tki = (const long long*)d_in[1];  // int64 indices
  const float* tkw = (const float*)d_in[2];
  const float* w1 = (const float*)d_in[3];
  const float* b1 = (const float*)d_in[4];
  const float* w2 = (const float*)d_in[5];
  const float* b2 = (const float*)d_in[6];
  float* out = (float*)d_out;

  // workspace layout: [counts: 16 ints][tok: E*CAP ints][twg: E*CAP floats]
  int* counts = (int*)d_ws;
  int* tok = counts + 16;
  float* twg = (float*)(tok + (size_t)E_ * CAP_);

  zero_kernel<<<2048, 256, 0, stream>>>(out, counts, (size_t)N_ * H_);
  route_kernel<<<(N_ + 255) / 256, 256, 0, stream>>>(tki, tkw, counts, tok, twg);

  const size_t smem = (size_t)TM_ * LDK_ * sizeof(float) + TM_ * 8;  // ~257.5 KB
  (void)hipFuncSetAttribute((const void*)moe_expert_gemm,
                            hipFuncAttributeMaxDynamicSharedMemorySize,
                            (int)smem);
  moe_expert_gemm<<<dim3(TILES_, E_), BLOCK_, smem, stream>>>(
      X, w1, b1, w2, b2, counts, tok, twg, out);
}